// Decoder_1675037245702
// MI455X (gfx1250) — compile-verified
//
#include <hip/hip_runtime.h>
#include <math.h>

// ---------------------------------------------------------------------------
// CDNA5 (gfx1250) implementation of the ONERA GNN decoder.
// GEMMs use V_WMMA_F32_16X16X4_F32 (exact f32 matrix path, wave32).
// B operand staged pair-interleaved in LDS so each WMMA fragment is a single
// ds_load_b64; epilogue specialized at compile time (no runtime branches).
// All GEMM M values are multiples of 128 (16384/32768/65536), so the block
// tile is always full and no row guards are needed.
// ---------------------------------------------------------------------------

typedef float v2f __attribute__((ext_vector_type(2)));
typedef float v8f __attribute__((ext_vector_type(8)));

#define LDT    136  // LDS row stride (floats) for the A tile
#define KPMAX  132  // max padded K (131 -> 132)
#define NPAIR  (KPMAX / 2)   // 66 K-pairs
#define WPSTR  264  // floats per K-pair row in LDS (128 cols * 2 + 8 pad)

// -------------------------------- utility ----------------------------------

__global__ void fill_zero_kernel(float* __restrict__ p, int n) {
  int i = blockIdx.x * 256 + threadIdx.x;
  if (i < n) p[i] = 0.f;
}

__global__ void transform_kernel(const float* __restrict__ pos, int N,
                                 float* __restrict__ tp) {
  int i = blockIdx.x * 256 + threadIdx.x;
  if (i >= N) return;
  float x = pos[i * 3 + 0], y = pos[i * 3 + 1], z = pos[i * 3 + 2];
  float nx = x - 0.57735026918962576f * y;            // tan(pi/6)
  float s  = 1.f + 0.78571428571428570f * (y / 1.1963f);
  tp[i * 3 + 0] = nx * s;
  tp[i * 3 + 1] = y * s;
  tp[i * 3 + 2] = z * s;
}

// xc[M,132] = concat(x[M,128], pos[M,3], 0-pad)
__global__ void concat_kernel(const float* __restrict__ x,
                              const float* __restrict__ pos, int M,
                              float* __restrict__ xc) {
  int i = blockIdx.x * 256 + threadIdx.x;
  if (i >= M * 132) return;
  int r = i / 132, c = i % 132;
  float v;
  if (c < 128)      v = x[(size_t)r * 128 + c];
  else if (c < 131) v = pos[(size_t)r * 3 + (c - 128)];
  else              v = 0.f;
  xc[i] = v;
}

// ------------------------------ WMMA GEMM ----------------------------------
// C[M,128] = A[M,K] @ W[K,128] (+bias) (+agg).  256 threads = 8 wave32s.
// Block tile: 128 rows x 128 cols (M % 128 == 0).  Wave w: rows [16w,16w+16).
template <bool HAS_BIAS, bool HAS_AGG>
__global__ void gemm_wmma_kernel(const float* __restrict__ A, int lda, int M,
                                 int K, const float* __restrict__ W,
                                 const float* __restrict__ bias,
                                 const float* __restrict__ agg,
                                 float* __restrict__ C) {
  extern __shared__ float smem[];
  float* sA = smem;                 // [128][LDT]
  float* sW = smem + 128 * LDT;     // [NPAIR][WPSTR], pair-interleaved cols

  const int tid = threadIdx.x;
  const int block_row = blockIdx.x * 128;

  // Stage W pair-interleaved: sW[p][2n] = W[2p][n], sW[p][2n+1] = W[2p+1][n]
  for (int i = tid; i < NPAIR * 128; i += 256) {
    int p = i >> 7, n = i & 127;
    int k = 2 * p;
    float w0 = (k < K)     ? W[(size_t)k * 128 + n]       : 0.f;
    float w1 = (k + 1 < K) ? W[(size_t)(k + 1) * 128 + n] : 0.f;
    v2f wv; wv.x = w0; wv.y = w1;
    *(v2f*)&sW[p * WPSTR + n * 2] = wv;
  }
  // Stage A tile (coalesced, zero-padded to KPMAX cols)
  for (int i = tid; i < 128 * KPMAX; i += 256) {
    int r = i / KPMAX, k = i % KPMAX;
    int gr = block_row + r;
    sA[r * LDT + k] = (k < K) ? A[(size_t)gr * lda + k] : 0.f;
  }
  __syncthreads();

  const int wave = tid >> 5;
  const int lane = tid & 31;
  const int half = lane >> 4;   // 0: K pair {0,1}, 1: K pair {2,3}
  const int l16  = lane & 15;
  const int arow = wave * 16 + l16;

  v8f acc[8] = {};

  const int Kq = (K + 3) & ~3;
  for (int kk = 0; kk < Kq; kk += 4) {
    const int k0 = kk + half * 2;          // even -> 8B-aligned LDS access
    const int p0 = (kk >> 1) + half;       // K-pair index for this half-wave
    const v2f a = *(const v2f*)&sA[arow * LDT + k0];
    const float* wrow = &sW[p0 * WPSTR + l16 * 2];
#pragma unroll
    for (int t = 0; t < 8; ++t) {
      const v2f b = *(const v2f*)&wrow[t * 32];   // 16 cols * 2 floats
      acc[t] = __builtin_amdgcn_wmma_f32_16x16x4_f32(
          /*neg_a=*/false, a, /*neg_b=*/false, b,
          /*c_mod=*/(short)0, acc[t], /*reuse_a=*/false, /*reuse_b=*/false);
    }
  }

  // C/D layout: VGPR v -> M = v + 8*half, N = l16 (+16t)
#pragma unroll
  for (int t = 0; t < 8; ++t) {
    const int col = t * 16 + l16;
    float bv = 0.f;
    if (HAS_BIAS) bv = bias[col];
#pragma unroll
    for (int v = 0; v < 8; ++v) {
      const int r = block_row + wave * 16 + v + half * 8;
      float val = acc[t][v] + bv;
      if (HAS_AGG) val += agg[(size_t)r * 128 + col];
      C[(size_t)r * 128 + col] = val;
    }
  }
}

// ------------------------------ BatchNorm ----------------------------------

__global__ void bn_stats_kernel(const float* __restrict__ y, int M,
                                float* __restrict__ stats /*[256]*/) {
  int c = threadIdx.x;              // 128 threads
  int r0 = blockIdx.x * 512;
  int r1 = r0 + 512; if (r1 > M) r1 = M;
  float s = 0.f, s2 = 0.f;
  for (int r = r0; r < r1; ++r) {
    float v = y[(size_t)r * 128 + c];
    s += v; s2 += v * v;
  }
  atomicAdd(&stats[c], s);
  atomicAdd(&stats[128 + c], s2);
}

__global__ void bn_elu_kernel(const float* __restrict__ y, int M,
                              const float* __restrict__ stats,
                              const float* __restrict__ g,
                              const float* __restrict__ b,
                              float* __restrict__ x) {
  int i = blockIdx.x * 256 + threadIdx.x;
  if (i >= M * 128) return;
  int c = i & 127;
  float inv  = 1.f / (float)M;
  float mean = stats[c] * inv;
  float var  = stats[128 + c] * inv - mean * mean;
  float v = g[c] * (y[i] - mean) * rsqrtf(var + 1e-5f) + b[c];
  x[i] = v > 0.f ? v : expm1f(v);
}

// ---------------------------- edge scatter-add -----------------------------

__global__ void scatter_add_kernel(const float* __restrict__ h,
                                   const int* __restrict__ src,
                                   const int* __restrict__ dst, int E,
                                   float* __restrict__ agg) {
  int i = blockIdx.x * 256 + threadIdx.x;
  if (i >= E * 32) return;
  int e = i >> 5;
  int f = (i & 31) * 4;
  int s = src[e], d = dst[e];
  const float* hp = h + (size_t)s * 128 + f;
  float* ap = agg + (size_t)d * 128 + f;
  atomicAdd(ap + 0, hp[0]);
  atomicAdd(ap + 1, hp[1]);
  atomicAdd(ap + 2, hp[2]);
  atomicAdd(ap + 3, hp[3]);
}

// --------------------------------- KNN -------------------------------------

#define KTILE 1024
__global__ void knn_kernel(const float* __restrict__ qp, int Ny,
                           const float* __restrict__ sp, int Nx,
                           float* __restrict__ kd, int* __restrict__ ki) {
  __shared__ float st[KTILE * 3];
  int q = blockIdx.x * 256 + threadIdx.x;
  float qx = 0.f, qy = 0.f, qz = 0.f;
  if (q < Ny) { qx = qp[q * 3]; qy = qp[q * 3 + 1]; qz = qp[q * 3 + 2]; }
  float d0 = 3.4e38f, d1 = 3.4e38f, d2 = 3.4e38f;
  int   i0 = 0, i1 = 0, i2 = 0;
  for (int base = 0; base < Nx; base += KTILE) {
    int lim = Nx - base; if (lim > KTILE) lim = KTILE;
    for (int i = threadIdx.x; i < lim * 3; i += 256)
      st[i] = sp[(size_t)base * 3 + i];
    __syncthreads();
    for (int j = 0; j < lim; ++j) {
      float dx = qx - st[j * 3], dy = qy - st[j * 3 + 1], dz = qz - st[j * 3 + 2];
      float d = dx * dx + dy * dy + dz * dz;
      int gi = base + j;
      if (d < d2) {
        if (d < d1) {
          if (d < d0) { d2 = d1; i2 = i1; d1 = d0; i1 = i0; d0 = d; i0 = gi; }
          else        { d2 = d1; i2 = i1; d1 = d;  i1 = gi; }
        } else        { d2 = d;  i2 = gi; }
      }
    }
    __syncthreads();
  }
  if (q < Ny) {
    kd[q * 3] = d0; kd[q * 3 + 1] = d1; kd[q * 3 + 2] = d2;
    ki[q * 3] = i0; ki[q * 3 + 1] = i1; ki[q * 3 + 2] = i2;
  }
}

__global__ void interp_kernel(const float* __restrict__ f,
                              const float* __restrict__ kd,
                              const int* __restrict__ ki, int Ny,
                              float* __restrict__ out) {
  int i = blockIdx.x * 256 + threadIdx.x;
  if (i >= Ny * 128) return;
  int q = i >> 7, c = i & 127;
  float w0 = 1.f / fmaxf(kd[q * 3 + 0], 1e-16f);
  float w1 = 1.f / fmaxf(kd[q * 3 + 1], 1e-16f);
  float w2 = 1.f / fmaxf(kd[q * 3 + 2], 1e-16f);
  float s = w0 * f[(size_t)ki[q * 3 + 0] * 128 + c] +
            w1 * f[(size_t)ki[q * 3 + 1] * 128 + c] +
            w2 * f[(size_t)ki[q * 3 + 2] * 128 + c];
  out[i] = s / (w0 + w1 + w2);
}

// --------------------------- OUT=3 output layer ----------------------------

__global__ void gemv3_kernel(const float* __restrict__ xc, int M,
                             const float* __restrict__ W /*[131,3]*/,
                             float* __restrict__ h3) {
  int i = blockIdx.x * 256 + threadIdx.x;
  if (i >= M * 3) return;
  int r = i / 3, o = i % 3;
  const float* a = xc + (size_t)r * 132;
  float s = 0.f;
  for (int k = 0; k < 131; ++k) s += a[k] * W[k * 3 + o];
  h3[i] = s;
}

__global__ void scatter3_kernel(const float* __restrict__ h3,
                                const int* __restrict__ src,
                                const int* __restrict__ dst, int E,
                                float* __restrict__ agg3) {
  int i = blockIdx.x * 256 + threadIdx.x;
  if (i >= E * 3) return;
  int e = i / 3, o = i % 3;
  atomicAdd(&agg3[(size_t)dst[e] * 3 + o], h3[(size_t)src[e] * 3 + o]);
}

__global__ void final3_kernel(const float* __restrict__ xc, int M,
                              const float* __restrict__ Wr /*[131,3]*/,
                              const float* __restrict__ b,
                              const float* __restrict__ agg3,
                              float* __restrict__ out) {
  int i = blockIdx.x * 256 + threadIdx.x;
  if (i >= M * 3) return;
  int r = i / 3, o = i % 3;
  const float* a = xc + (size_t)r * 132;
  float s = 0.f;
  for (int k = 0; k < 131; ++k) s += a[k] * Wr[k * 3 + o];
  out[i] = s + agg3[i] + b[o];
}

// ------------------------------- host side ---------------------------------

static inline int cdiv_i(int a, int b) { return (a + b - 1) / b; }

extern "C" void kernel_launch(void* const* d_in, const int* in_sizes, int n_in,
                              void* d_out, int out_size, void* d_ws,
                              size_t ws_size, hipStream_t stream) {
  const float* latent = (const float*)d_in[0];
  const float* pos0   = (const float*)d_in[1];
  const float* pos1   = (const float*)d_in[2];
  const float* pos2   = (const float*)d_in[3];
  const float* lin_W  = (const float*)d_in[4];
  const float* lin_b  = (const float*)d_in[5];
  const float* bn_g   = (const float*)d_in[6];   // [5,128]
  const float* bn_b   = (const float*)d_in[7];   // [5,128]
  const float* Wr_h   = (const float*)d_in[8];   // [4,131,128]
  const float* Wn_h   = (const float*)d_in[9];   // [4,131,128]
  const float* b_h    = (const float*)d_in[10];  // [4,128]
  const float* Wr_o   = (const float*)d_in[11];  // [131,3]
  const float* Wn_o   = (const float*)d_in[12];  // [131,3]
  const float* b_o    = (const float*)d_in[13];  // [3]
  const int*   ei0    = (const int*)d_in[14];
  const int*   ei1    = (const int*)d_in[15];
  const int*   ei2    = (const int*)d_in[16];
  float* out = (float*)d_out;

  const int N0 = in_sizes[1] / 3, N1 = in_sizes[2] / 3, N2 = in_sizes[3] / 3;
  const int E0 = in_sizes[14] / 2, E1 = in_sizes[15] / 2, E2 = in_sizes[16] / 2;
  const int NMAX = N2;

  // ---- bump allocator over workspace ----
  char* wp = (char*)d_ws;
  auto alloc = [&](size_t bytes) -> void* {
    void* r = (void*)wp;
    wp += (bytes + 255) & ~(size_t)255;
    return r;
  };
  float* buf_x  = (float*)alloc((size_t)NMAX * 128 * 4);
  float* buf_xB = (float*)alloc((size_t)NMAX * 128 * 4);
  float* buf_xc = (float*)alloc((size_t)NMAX * 132 * 4);
  float* buf_hn = (float*)alloc((size_t)NMAX * 128 * 4);
  float* buf_agg= (float*)alloc((size_t)NMAX * 128 * 4);
  float* stats  = (float*)alloc(256 * 4);
  float* tp0    = (float*)alloc((size_t)N0 * 3 * 4);
  float* tp1    = (float*)alloc((size_t)N1 * 3 * 4);
  float* tp2    = (float*)alloc((size_t)N2 * 3 * 4);
  float* knn_d  = (float*)alloc((size_t)NMAX * 3 * 4);
  int*   knn_i  = (int*)  alloc((size_t)NMAX * 3 * 4);
  float* agg3   = (float*)alloc((size_t)NMAX * 3 * 4);

  const size_t GEMM_LDS =
      ((size_t)128 * LDT + (size_t)NPAIR * WPSTR) * sizeof(float);

  auto zero = [&](float* p, int n) {
    fill_zero_kernel<<<cdiv_i(n, 256), 256, 0, stream>>>(p, n);
  };
  auto bnstats = [&](const float* y, int M) {
    zero(stats, 256);
    bn_stats_kernel<<<cdiv_i(M, 512), 128, 0, stream>>>(y, M, stats);
  };
  auto bnelu = [&](const float* y, int M, int layer, float* x) {
    bn_elu_kernel<<<cdiv_i(M * 128, 256), 256, 0, stream>>>(
        y, M, stats, bn_g + (size_t)layer * 128, bn_b + (size_t)layer * 128, x);
  };

  // graph_conv + BN + ELU
  auto conv = [&](const float* x_in, float* x_out, const float* pos, int M,
                  const int* ei, int E, int wlayer, int bnlayer) {
    const float* Wr = Wr_h + (size_t)wlayer * 131 * 128;
    const float* Wn = Wn_h + (size_t)wlayer * 131 * 128;
    const float* bh = b_h + (size_t)wlayer * 128;
    concat_kernel<<<cdiv_i(M * 132, 256), 256, 0, stream>>>(x_in, pos, M, buf_xc);
    gemm_wmma_kernel<false, false><<<M / 128, 256, GEMM_LDS, stream>>>(
        buf_xc, 132, M, 131, Wn, nullptr, nullptr, buf_hn);
    zero(buf_agg, M * 128);
    scatter_add_kernel<<<cdiv_i(E * 32, 256), 256, 0, stream>>>(
        buf_hn, ei /*src*/, ei + E /*dst*/, E, buf_agg);
    gemm_wmma_kernel<true, true><<<M / 128, 256, GEMM_LDS, stream>>>(
        buf_xc, 132, M, 131, Wr, bh, buf_agg, buf_hn);  // y (pre-BN)
    bnstats(buf_hn, M);
    bnelu(buf_hn, M, bnlayer, x_out);
  };

  // ---- transformed positions for KNN ----
  transform_kernel<<<cdiv_i(N0, 256), 256, 0, stream>>>(pos0, N0, tp0);
  transform_kernel<<<cdiv_i(N1, 256), 256, 0, stream>>>(pos1, N1, tp1);
  transform_kernel<<<cdiv_i(N2, 256), 256, 0, stream>>>(pos2, N2, tp2);

  // ---- input linear + BN + ELU ----
  gemm_wmma_kernel<true, false><<<N0 / 128, 256, GEMM_LDS, stream>>>(
      latent, 64, N0, 64, lin_W, lin_b, nullptr, buf_hn);
  bnstats(buf_hn, N0);
  bnelu(buf_hn, N0, 0, buf_x);

  // ---- two convs at level 0 ----
  conv(buf_x, buf_xB, pos0, N0, ei0, E0, 0, 1);
  conv(buf_xB, buf_x, pos0, N0, ei0, E0, 1, 2);

  // ---- pool 0 -> 1 ----
  knn_kernel<<<cdiv_i(N1, 256), 256, 0, stream>>>(tp1, N1, tp0, N0, knn_d, knn_i);
  interp_kernel<<<cdiv_i(N1 * 128, 256), 256, 0, stream>>>(buf_x, knn_d, knn_i,
                                                           N1, buf_xB);
  conv(buf_xB, buf_x, pos1, N1, ei1, E1, 2, 3);

  // ---- pool 1 -> 2 ----
  knn_kernel<<<cdiv_i(N2, 256), 256, 0, stream>>>(tp2, N2, tp1, N1, knn_d, knn_i);
  interp_kernel<<<cdiv_i(N2 * 128, 256), 256, 0, stream>>>(buf_x, knn_d, knn_i,
                                                           N2, buf_xB);
  conv(buf_xB, buf_x, pos2, N2, ei2, E2, 3, 4);

  // ---- OUT=3 output GraphConv (no BN/ELU) ----
  concat_kernel<<<cdiv_i(N2 * 132, 256), 256, 0, stream>>>(buf_x, pos2, N2, buf_xc);
  gemv3_kernel<<<cdiv_i(N2 * 3, 256), 256, 0, stream>>>(buf_xc, N2, Wn_o, buf_hn);
  zero(agg3, N2 * 3);
  scatter3_kernel<<<cdiv_i(E2 * 3, 256), 256, 0, stream>>>(buf_hn, ei2, ei2 + E2,
                                                           E2, agg3);
  final3_kernel<<<cdiv_i(N2 * 3, 256), 256, 0, stream>>>(buf_xc, N2, Wr_o, b_o,
                                                         agg3, out);
}